// AutoRec_24223615550487
// MI455X (gfx1250) — compile-verified
//
#include <hip/hip_runtime.h>
#include <hip/hip_bf16.h>

typedef __attribute__((ext_vector_type(2))) float v2f;
typedef __attribute__((ext_vector_type(8))) float v8f;

// ---------------------------------------------------------------------------
// Kernel 0: zero agg workspace (M*D floats) + the global scalar accumulator.
// ---------------------------------------------------------------------------
__global__ void ar_zero_k(float* __restrict__ p, size_t total) {
    size_t t = (size_t)blockIdx.x * blockDim.x + threadIdx.x;
    if (t < total) p[t] = 0.0f;
}

// ---------------------------------------------------------------------------
// Kernel 1: COO SpMM scatter. One wave per nonzero: 32 lanes each load a
// float4 of v[col] (coalesced 512B row read) and scatter-add into agg[row].
// ---------------------------------------------------------------------------
__global__ void ar_spmm_k(const int* __restrict__ ij, const float* __restrict__ r,
                          const float* __restrict__ v, float* __restrict__ agg,
                          int nnz, int D) {
    const int lane = threadIdx.x & 31;
    const int k = blockIdx.x * (blockDim.x >> 5) + (threadIdx.x >> 5);
    if (k >= nnz) return;
    const int row = ij[k];         // ij[0][k]
    const int col = ij[nnz + k];   // ij[1][k]
    const float rv = r[k];
    const float* __restrict__ vrow = v + (size_t)col * D;
    float* __restrict__ arow = agg + (size_t)row * D;
    for (int d = lane * 4; d < D; d += 128) {   // D % 4 == 0 (D = 128)
        float4 x = *(const float4*)(vrow + d);
        atomicAdd(arow + d + 0, rv * x.x);
        atomicAdd(arow + d + 1, rv * x.y);
        atomicAdd(arow + d + 2, rv * x.z);
        atomicAdd(arow + d + 3, rv * x.w);
    }
}

// ---------------------------------------------------------------------------
// Kernel 2: h = sigmoid(agg + mu), in place. blockDim.x == D, grid == M.
// ---------------------------------------------------------------------------
__global__ void ar_sigmoid_k(float* __restrict__ agg, const float* __restrict__ mu) {
    const int d = threadIdx.x;
    const size_t idx = (size_t)blockIdx.x * blockDim.x + d;
    const float x = agg[idx] + mu[d];
    agg[idx] = 1.0f / (1.0f + __expf(-x));
}

// ---------------------------------------------------------------------------
// Kernel 3: scalar = sum_e dot(h[i[e]], w[j[e]]) via WMMA diagonal trick.
//
// Per wave: process groups of 16 pairs. A (16xK) rows = gathered h rows,
// B (KxC=16) cols = gathered w rows. V_WMMA_F32_16X16X4_F32 accumulates
// C += A*B across ALL groups handled by this wave; diag(C) accumulates the
// wanted dot products, off-diagonal garbage is discarded. One atomicAdd per
// wave at the end.
//
// f32 16x16x4 fragment layouts (ISA 7.12.2):
//   A: lane L holds row M=L%16; VGPR {0,1} = K {0,1} (lanes 0-15) or
//      K {2,3} (lanes 16-31)  -> float2 at hrow[kk + 2*(L/16)]
//   B: mirrored: lane L holds col N=L%16, K split the same way
//      -> float2 at wrow[kk + 2*(L/16)]
//   C (16x16 f32): VGPR g: lanes 0-15 -> (M=g,   N=lane),
//                          lanes 16-31 -> (M=8+g, N=lane-16)
//      diag(m=g)   at VGPR g, lane g
//      diag(m=8+g) at VGPR g, lane 24+g
// ---------------------------------------------------------------------------
__global__ void ar_dot_wmma_k(const float* __restrict__ h, const float* __restrict__ w,
                              const int* __restrict__ iv, const int* __restrict__ jv,
                              int groups, int D, float* __restrict__ scal) {
    const int lane = threadIdx.x & 31;
    const int n    = lane & 15;   // pair-within-group (= A row = B col)
    const int half = lane >> 4;   // which K-pair this lane supplies
    const int wavesPerBlock = blockDim.x >> 5;
    const int gwave  = blockIdx.x * wavesPerBlock + (threadIdx.x >> 5);
    const int nwaves = gridDim.x * wavesPerBlock;

    v8f c = {};  // accumulates across all groups handled by this wave

    for (int g = gwave; g < groups; g += nwaves) {   // wave-uniform loop
        const int e = g * 16 + n;
        const float* __restrict__ hrow = h + (size_t)iv[e] * D;
        const float* __restrict__ wrow = w + (size_t)jv[e] * D;
#pragma unroll 8
        for (int kk = 0; kk < D; kk += 4) {
            const int off = kk + 2 * half;
            v2f a  = *(const v2f*)(hrow + off);
            v2f bb = *(const v2f*)(wrow + off);
            c = __builtin_amdgcn_wmma_f32_16x16x4_f32(
                    /*neg_a=*/false, a, /*neg_b=*/false, bb,
                    /*c_mod=*/(short)0, c, /*reuse_a=*/false, /*reuse_b=*/false);
        }
    }

    // Pull this lane's diagonal contribution out of the C fragment.
    const int idx = (lane < 8) ? lane : ((lane >= 24) ? (lane - 24) : -1);
    float dsum = 0.0f;
    dsum = (idx == 0) ? c[0] : dsum;
    dsum = (idx == 1) ? c[1] : dsum;
    dsum = (idx == 2) ? c[2] : dsum;
    dsum = (idx == 3) ? c[3] : dsum;
    dsum = (idx == 4) ? c[4] : dsum;
    dsum = (idx == 5) ? c[5] : dsum;
    dsum = (idx == 6) ? c[6] : dsum;
    dsum = (idx == 7) ? c[7] : dsum;

    for (int off = 16; off > 0; off >>= 1)
        dsum += __shfl_xor(dsum, off, 32);
    if (lane == 0)
        atomicAdd(scal, dsum);
}

// ---------------------------------------------------------------------------
// Kernel 3b: scalar tail for E % 16 != 0 (not hit for E = 1e6, kept generic).
// ---------------------------------------------------------------------------
__global__ void ar_dot_tail_k(const float* __restrict__ h, const float* __restrict__ w,
                              const int* __restrict__ iv, const int* __restrict__ jv,
                              int start, int E, int D, float* __restrict__ scal) {
    const int lane = threadIdx.x;  // one wave
    float acc = 0.0f;
    for (int e = start; e < E; ++e) {
        const float* hrow = h + (size_t)iv[e] * D;
        const float* wrow = w + (size_t)jv[e] * D;
        for (int d = lane; d < D; d += 32) acc += hrow[d] * wrow[d];
    }
    for (int off = 16; off > 0; off >>= 1) acc += __shfl_xor(acc, off, 32);
    if (lane == 0) atomicAdd(scal, acc);
}

// ---------------------------------------------------------------------------
// Kernel 4: out[e] = scalar + b[j[e]].
// ---------------------------------------------------------------------------
__global__ void ar_finalize_k(const float* __restrict__ scal, const float* __restrict__ b,
                              const int* __restrict__ jv, float* __restrict__ out, int E) {
    const int e = blockIdx.x * blockDim.x + threadIdx.x;
    if (e < E) out[e] = *scal + b[jv[e]];
}

extern "C" void kernel_launch(void* const* d_in, const int* in_sizes, int n_in,
                              void* d_out, int out_size, void* d_ws, size_t ws_size,
                              hipStream_t stream) {
    const int*   ij  = (const int*)d_in[0];    // (2, NNZ)
    const float* r   = (const float*)d_in[1];  // (NNZ,)
    // d_in[2] = m (device scalar; M derived below: M == N == len(b) here)
    const int*   iv  = (const int*)d_in[3];    // (E,)
    const int*   jv  = (const int*)d_in[4];    // (E,)
    const float* v   = (const float*)d_in[5];  // (N, D)
    const float* mu  = (const float*)d_in[6];  // (1, D)
    const float* w   = (const float*)d_in[7];  // (N, D)
    const float* b   = (const float*)d_in[8];  // (N,)
    float*       out = (float*)d_out;          // (E,)

    const int NNZ = in_sizes[1];
    const int E   = in_sizes[3];
    const int D   = in_sizes[6];               // 128
    const int M   = in_sizes[8];               // num_segments m == N in this problem

    float* agg  = (float*)d_ws;                // M*D floats; becomes h in place
    float* scal = agg + (size_t)M * D;         // 1 float global accumulator

    // 0) zero agg + scalar
    const size_t ztotal = (size_t)M * D + 1;
    const int zblocks = (int)((ztotal + 255) / 256);
    ar_zero_k<<<zblocks, 256, 0, stream>>>(agg, ztotal);

    // 1) COO SpMM scatter: one wave per nnz (256 threads = 8 waves / block)
    const int nnzBlocks = (NNZ + 7) / 8;
    ar_spmm_k<<<nnzBlocks, 256, 0, stream>>>(ij, r, v, agg, NNZ, D);

    // 2) h = sigmoid(agg + mu), in place (blockDim = D = 128 -> 4 waves)
    ar_sigmoid_k<<<M, D, 0, stream>>>(agg, mu);

    // 3) global pair-dot reduce via WMMA (one atomic per wave)
    const int groups = E / 16;
    if (groups > 0)
        ar_dot_wmma_k<<<256, 256, 0, stream>>>(agg, w, iv, jv, groups, D, scal);
    const int tail_start = groups * 16;
    if (tail_start < E)
        ar_dot_tail_k<<<1, 32, 0, stream>>>(agg, w, iv, jv, tail_start, E, D, scal);

    // 4) out[e] = scalar + b[j[e]]
    ar_finalize_k<<<(E + 255) / 256, 256, 0, stream>>>(scal, b, jv, out, E);
}